// CRF_1786706395822
// MI455X (gfx1250) — compile-verified
//
#include <hip/hip_runtime.h>
#include <hip/hip_bf16.h>
#include <cmath>

// CRF log-likelihood on gfx1250 (MI455X).
//   L=2048 timesteps, B=2048 sequences, T=7 tags.
//
// Denominator forward scan: exp-domain rewrite
//   S[b,j] = sum_i exp(alpha[b,i]-m[b]) * exp(trans[i,j])
// as a [16 x 8] @ [8 x 7] f32 matmul -> two chained V_WMMA_F32_16X16X4_F32,
// done for both self/other transition tables (4 WMMAs/step/wave) with a
// per-row contagion select. Next step's emission tile (448B/wave) is staged
// into double-buffered LDS with GLOBAL_LOAD_ASYNC_TO_LDS_B128 (ASYNCcnt);
// qmask/mask flags are software-pipelined one step ahead in registers, so the
// steady-state loop has no global-load waits on its critical path.
//
// Numerator: parallelized over L in 16 chunks (partial score + mask count),
// start/end terms resolved in the final deterministic reduction.

#define LLEN 2048
#define BATCH 2048
#define TAGS 7
#define WAVES 4            // waves per block in the forward kernel
#define CHUNKS 16          // L-chunks for the numerator
#define CLEN (LLEN / CHUNKS)

typedef __attribute__((ext_vector_type(2))) float v2f;
typedef __attribute__((ext_vector_type(8))) float v8f;

#define LOG2E 1.4426950408889634f
#define LN2 0.6931471805599453f

// Intra-wave LDS ordering (DS ops of one wave are in-order; wait completion
// so other lanes observe the writes).
#define WAVE_DS_FENCE() asm volatile("s_wait_dscnt 0" ::: "memory")

// ---------------------------------------------------------------------------
// Kernel 1: numerator partials. blockIdx.y = L-chunk, threads over batch.
// ---------------------------------------------------------------------------
__global__ void crf_score_partial_kernel(const float* __restrict__ em,
                                         const int* __restrict__ tags,
                                         const int* __restrict__ qmask,
                                         const int* __restrict__ mask,
                                         const float* __restrict__ selfT,
                                         const float* __restrict__ otherT,
                                         float* __restrict__ outPartial,
                                         int* __restrict__ outCnt) {
    __shared__ float sSelf[TAGS * TAGS];
    __shared__ float sOther[TAGS * TAGS];
    for (int i = threadIdx.x; i < TAGS * TAGS; i += blockDim.x) {
        sSelf[i] = selfT[i];
        sOther[i] = otherT[i];
    }
    __syncthreads();

    int b = blockIdx.x * blockDim.x + threadIdx.x;
    int ch = blockIdx.y;
    int l0 = ch * CLEN;
    int l1 = l0 + CLEN;

    float acc = 0.0f;
    int cnt = 0;
    int lstart = (l0 == 0) ? 1 : l0;
    if (l0 == 0) cnt += (mask[b] != 0);

    int prev = tags[(lstart - 1) * BATCH + b];
    int prevq = qmask[(lstart - 1) * BATCH + b];

    for (int l = lstart; l < l1; ++l) {
        int idx = l * BATCH + b;
        int t = tags[idx];
        int q = qmask[idx];
        int m = mask[idx];
        float tr = (q != prevq) ? sOther[prev * TAGS + t] : sSelf[prev * TAGS + t];
        float e = em[(size_t)idx * TAGS + t];
        acc += (tr + e) * (float)m;
        cnt += (m != 0);
        prev = t;
        prevq = q;
    }
    outPartial[ch * BATCH + b] = acc;
    outCnt[ch * BATCH + b] = cnt;
}

// ---------------------------------------------------------------------------
// Kernel 2: denominator (partition function) via WMMA-based forward scan.
// One wave owns 16 batch rows. C-tile layout: vgpr r, lane: M(batch)=r+8*hi,
// N(tag)=lane&15.  A-tile: lane row M=lane&15, K pair = 2*hi (+4 hi WMMA).
// ---------------------------------------------------------------------------
__global__ void __launch_bounds__(32 * WAVES)
crf_fwd_kernel(const float* __restrict__ em,
               const int* __restrict__ qmask,
               const int* __restrict__ mask,
               const float* __restrict__ startT,
               const float* __restrict__ endT,
               const float* __restrict__ selfT,
               const float* __restrict__ otherT,
               float* __restrict__ outLogZ) {
    __shared__ float sAlpha[WAVES][16][16];   // rows widened: all lanes store
    __shared__ float sE[WAVES][16][8];        // exp(alpha - max), K=7 pad = 0
    __shared__ float sMax[WAVES][16];
    __shared__ int sCM[WAVES][16];            // bit0 = contagion, bit1 = mask
    __shared__ float sEm[WAVES][2][128];      // double-buffered emission tile

    const int tid = threadIdx.x;
    const int wid = tid >> 5;
    const int lane = tid & 31;
    const int hi = lane >> 4;   // half-wave select
    const int lj = lane & 15;   // N (tag) in C layout / M (batch row) in A layout
    const int jj = (lj < TAGS) ? lj : 0;   // clamped tag column
    const int bbase = (blockIdx.x * WAVES + wid) * 16;

    // --- constant B fragments: exp(trans), K rows split across half-waves ---
    v2f bS0, bS1, bO0, bO1;
    {
        int k0 = 2 * hi, k1 = k0 + 1;
        bool c = (lj < TAGS);
        bS0.x = c ? __builtin_amdgcn_exp2f(selfT[k0 * TAGS + lj] * LOG2E) : 0.0f;
        bS0.y = c ? __builtin_amdgcn_exp2f(selfT[k1 * TAGS + lj] * LOG2E) : 0.0f;
        bO0.x = c ? __builtin_amdgcn_exp2f(otherT[k0 * TAGS + lj] * LOG2E) : 0.0f;
        bO0.y = c ? __builtin_amdgcn_exp2f(otherT[k1 * TAGS + lj] * LOG2E) : 0.0f;
        int k4 = k0 + 4, k5 = k4 + 1;   // k5 == 7 is the zero pad row
        bS1.x = (k4 < TAGS && c) ? __builtin_amdgcn_exp2f(selfT[k4 * TAGS + lj] * LOG2E) : 0.0f;
        bS1.y = (k5 < TAGS && c) ? __builtin_amdgcn_exp2f(selfT[k5 * TAGS + lj] * LOG2E) : 0.0f;
        bO1.x = (k4 < TAGS && c) ? __builtin_amdgcn_exp2f(otherT[k4 * TAGS + lj] * LOG2E) : 0.0f;
        bO1.y = (k5 < TAGS && c) ? __builtin_amdgcn_exp2f(otherT[k5 * TAGS + lj] * LOG2E) : 0.0f;
    }

    // --- alpha0 = start + emissions[0] (each lane owns (b=r+8*hi, j=lj)) ---
    float aOld[8];
    float st = startT[jj];
#pragma unroll
    for (int r = 0; r < 8; ++r) {
        int b = r + 8 * hi;
        float a = st + em[(size_t)(bbase + b) * TAGS + jj];
        if (lj >= TAGS) a = -1e30f;
        aOld[r] = a;
        sAlpha[wid][b][lj] = a;   // unconditional: rows are 16 wide
    }
    WAVE_DS_FENCE();

    // --- software-pipelined qmask/mask flags (lanes 0-15, b = lane) ---
    int qcur = 0, qnext = 0, mnext = 0;
    if (lane < 16) {
        int gb = bbase + lane;
        qcur = qmask[gb];            // l = 0
        qnext = qmask[BATCH + gb];   // l = 1
        mnext = mask[BATCH + gb];
    }

    // --- async-stage the l=1 emission tile (448B -> LDS buffer 0) ---
    {
        unsigned laddr = (unsigned)(uintptr_t)(&sEm[wid][0][0]) + 16u * (unsigned)lane;
        int gi = (lane < 28) ? lane : 27;   // clamp: lanes 28-31 fill LDS pad
        unsigned long long ga =
            (unsigned long long)(uintptr_t)(em + (size_t)(1 * BATCH + bbase) * TAGS) + 16u * gi;
        asm volatile("global_load_async_to_lds_b128 %0, %1, off"
                     :: "v"(laddr), "v"(ga) : "memory");
    }

    int buf = 0;
    for (int l = 1; l < LLEN; ++l) {
        // Wait for this step's emission tile, then launch next step's.
        asm volatile("s_wait_asynccnt 0x0" ::: "memory");
        if (l + 1 < LLEN) {
            unsigned laddr =
                (unsigned)(uintptr_t)(&sEm[wid][buf ^ 1][0]) + 16u * (unsigned)lane;
            int gi = (lane < 28) ? lane : 27;
            unsigned long long ga =
                (unsigned long long)(uintptr_t)(em + (size_t)((l + 1) * BATCH + bbase) * TAGS) +
                16u * gi;
            asm volatile("global_load_async_to_lds_b128 %0, %1, off"
                         :: "v"(laddr), "v"(ga) : "memory");
        }

        // Phase A: lanes 0-15 (b = lane): row max, exps, flags from pipeline.
        if (lane < 16) {
            int b = lane;
            const float4* ap = (const float4*)&sAlpha[wid][b][0];
            float4 q0 = ap[0];
            float4 q1 = ap[1];
            float mx = fmaxf(fmaxf(fmaxf(q0.x, q0.y), fmaxf(q0.z, q0.w)),
                             fmaxf(fmaxf(q1.x, q1.y), q1.z));
            float4 e0, e1;
            e0.x = __builtin_amdgcn_exp2f((q0.x - mx) * LOG2E);
            e0.y = __builtin_amdgcn_exp2f((q0.y - mx) * LOG2E);
            e0.z = __builtin_amdgcn_exp2f((q0.z - mx) * LOG2E);
            e0.w = __builtin_amdgcn_exp2f((q0.w - mx) * LOG2E);
            e1.x = __builtin_amdgcn_exp2f((q1.x - mx) * LOG2E);
            e1.y = __builtin_amdgcn_exp2f((q1.y - mx) * LOG2E);
            e1.z = __builtin_amdgcn_exp2f((q1.z - mx) * LOG2E);
            e1.w = 0.0f;   // K=7 pad column of A
            float4* ep = (float4*)&sE[wid][b][0];
            ep[0] = e0;
            ep[1] = e1;
            sMax[wid][b] = mx;
            sCM[wid][b] = ((qnext != qcur) ? 1 : 0) | (mnext ? 2 : 0);
            qcur = qnext;
            if (l + 1 < LLEN) {   // issue next step's flag loads (consumed next iter)
                int gb = bbase + b;
                qnext = qmask[(l + 1) * BATCH + gb];
                mnext = mask[(l + 1) * BATCH + gb];
            }
        }
        WAVE_DS_FENCE();

        // Build A fragments (lane: M=lj, K pair = 2*hi / 2*hi+4).
        int k0 = 2 * hi;
        v2f aLo, aHi;
        aLo.x = sE[wid][lj][k0 + 0];
        aLo.y = sE[wid][lj][k0 + 1];
        aHi.x = sE[wid][lj][k0 + 4];
        aHi.y = sE[wid][lj][k0 + 5];

        // S = exp(alpha-m) @ exp(trans), for self and other transition tables.
        v8f cS = {};
        v8f cO = {};
        cS = __builtin_amdgcn_wmma_f32_16x16x4_f32(false, aLo, false, bS0,
                                                   (short)0, cS, false, false);
        cS = __builtin_amdgcn_wmma_f32_16x16x4_f32(false, aHi, false, bS1,
                                                   (short)0, cS, false, false);
        cO = __builtin_amdgcn_wmma_f32_16x16x4_f32(false, aLo, false, bO0,
                                                   (short)0, cO, false, false);
        cO = __builtin_amdgcn_wmma_f32_16x16x4_f32(false, aHi, false, bO1,
                                                   (short)0, cO, false, false);

        // Per-half-wave metadata as vector LDS loads (broadcast).
        const int4* cmp = (const int4*)&sCM[wid][8 * hi];
        int4 c0 = cmp[0], c1 = cmp[1];
        const float4* mbp = (const float4*)&sMax[wid][8 * hi];
        float4 m0 = mbp[0], m1 = mbp[1];
        int cmArr[8] = {c0.x, c0.y, c0.z, c0.w, c1.x, c1.y, c1.z, c1.w};
        float mbArr[8] = {m0.x, m0.y, m0.z, m0.w, m1.x, m1.y, m1.z, m1.w};

        // Phase C: branch-free blend. S clamped so log stays finite even in
        // dead pad columns (keeps the 0*inf -> NaN case out of the blend).
        const float* emb = &sEm[wid][buf][0];
#pragma unroll
        for (int r = 0; r < 8; ++r) {
            int b = r + 8 * hi;
            int cm = cmArr[r];
            float S = (cm & 1) ? cO[r] : cS[r];
            S = fmaxf(S, 1e-37f);
            float emv = emb[b * TAGS + jj];
            float cand = mbArr[r] + __builtin_amdgcn_logf(S) * LN2 + emv;
            float sel = (cm & 2) ? 1.0f : 0.0f;
            float aN = fmaf(sel, cand - aOld[r], aOld[r]);
            aOld[r] = aN;
            sAlpha[wid][b][lj] = aN;   // unconditional store
        }
        WAVE_DS_FENCE();

        buf ^= 1;
    }

    // logZ[b] = logsumexp_j(alpha[b,j] + end[j])
    if (lane < 16) {
        int b = lane;
        float v[TAGS];
        float mx = -1e30f;
#pragma unroll
        for (int j = 0; j < TAGS; ++j) {
            v[j] = sAlpha[wid][b][j] + endT[j];
            mx = fmaxf(mx, v[j]);
        }
        float s = 0.0f;
#pragma unroll
        for (int j = 0; j < TAGS; ++j)
            s += __builtin_amdgcn_exp2f((v[j] - mx) * LOG2E);
        outLogZ[bbase + b] = mx + __builtin_amdgcn_logf(s) * LN2;
    }
}

// ---------------------------------------------------------------------------
// Kernel 3: deterministic final reduction.
// ---------------------------------------------------------------------------
__global__ void crf_final_kernel(const float* __restrict__ partial,
                                 const int* __restrict__ cnt,
                                 const float* __restrict__ logZ,
                                 const float* __restrict__ em,
                                 const int* __restrict__ tags,
                                 const float* __restrict__ startT,
                                 const float* __restrict__ endT,
                                 float* __restrict__ out) {
    __shared__ float s[256];
    float acc = 0.0f;
    for (int b = threadIdx.x; b < BATCH; b += 256) {
        int t0 = tags[b];
        float score = startT[t0] + em[(size_t)b * TAGS + t0];
        int c = 0;
        for (int ch = 0; ch < CHUNKS; ++ch) {
            score += partial[ch * BATCH + b];
            c += cnt[ch * BATCH + b];
        }
        int seqEnd = c - 1;
        score += endT[tags[seqEnd * BATCH + b]];
        acc += score - logZ[b];
    }
    s[threadIdx.x] = acc;
    __syncthreads();
    for (int off = 128; off > 0; off >>= 1) {
        if (threadIdx.x < off) s[threadIdx.x] += s[threadIdx.x + off];
        __syncthreads();
    }
    if (threadIdx.x == 0) out[0] = s[0];
}

// ---------------------------------------------------------------------------
extern "C" void kernel_launch(void* const* d_in, const int* in_sizes, int n_in,
                              void* d_out, int out_size, void* d_ws, size_t ws_size,
                              hipStream_t stream) {
    const float* em = (const float*)d_in[0];
    const int* tags = (const int*)d_in[1];
    const int* qmask = (const int*)d_in[2];
    const int* mask = (const int*)d_in[3];
    const float* startT = (const float*)d_in[4];
    const float* endT = (const float*)d_in[5];
    const float* selfT = (const float*)d_in[6];
    const float* otherT = (const float*)d_in[7];

    float* wsPartial = (float*)d_ws;                   // [CHUNKS*BATCH]
    int* wsCnt = (int*)(wsPartial + CHUNKS * BATCH);   // [CHUNKS*BATCH]
    float* wsLogZ = (float*)(wsCnt + CHUNKS * BATCH);  // [BATCH]
    float* outF = (float*)d_out;

    dim3 gScore(BATCH / 256, CHUNKS);
    crf_score_partial_kernel<<<gScore, 256, 0, stream>>>(
        em, tags, qmask, mask, selfT, otherT, wsPartial, wsCnt);

    crf_fwd_kernel<<<BATCH / (16 * WAVES), 32 * WAVES, 0, stream>>>(
        em, qmask, mask, startT, endT, selfT, otherT, wsLogZ);

    crf_final_kernel<<<1, 256, 0, stream>>>(
        wsPartial, wsCnt, wsLogZ, em, tags, startT, endT, outF);
}